// MSCLoss_90168543412886
// MI455X (gfx1250) — compile-verified
//
#include <hip/hip_runtime.h>
#include <math.h>

#define NN 32
#define DD 2048
#define K_TOP 5
#define UNKC 10
#define NU1 8
#define NU2 5
#define MR 3
#define MU_ 20
#define NP (NN - NU2)   // 27 picked columns
#define FC (MU_ + NU2)  // 25 final columns

typedef __attribute__((ext_vector_type(2))) float v2f;
typedef __attribute__((ext_vector_type(8))) float v8f;

// ---------------- Kernel 1: squared row norms (rows 0..31 = src, 32..63 = tgt)
__global__ void msc_norms(const float* __restrict__ src,
                          const float* __restrict__ tgt,
                          float* __restrict__ n2) {
  __shared__ float red[256];
  int r = blockIdx.x;
  const float* base = (r < NN) ? (src + (size_t)r * DD) : (tgt + (size_t)(r - NN) * DD);
  float acc = 0.f;
  for (int i = threadIdx.x; i < DD; i += 256) {
    float v = base[i];
    acc = fmaf(v, v, acc);
  }
  red[threadIdx.x] = acc;
  __syncthreads();
  for (int s = 128; s > 0; s >>= 1) {
    if ((int)threadIdx.x < s) red[threadIdx.x] += red[threadIdx.x + s];
    __syncthreads();
  }
  if (threadIdx.x == 0) n2[r] = red[0];
}

// ---------------- Kernel 2: sim = 1/(dist+1) via f32 WMMA Gram matrix
// One wave32 per 16x16 tile of the 32x32 sim matrix. d2 = |s|^2+|t|^2-2*s.t
__global__ void msc_sim_wmma(const float* __restrict__ src,
                             const float* __restrict__ tgt,
                             const float* __restrict__ n2,
                             float* __restrict__ sim) {
  int lane = threadIdx.x & 31;
  int half = lane >> 4;   // 0 or 1
  int lm   = lane & 15;
  int mi = blockIdx.x >> 1;   // src tile
  int ni = blockIdx.x & 1;    // tgt tile

  // A (16x4 f32): lane holds M=lane%16, VGPR v holds K = 2*(lane/16)+v
  // B (4x16 f32): lane holds N=lane%16, VGPR v holds K = 2*(lane/16)+v
  //   B[k][n] = tgt[n][k] -> same contiguous float2 load pattern from tgt row.
  const float* arow = src + (size_t)(mi * 16 + lm) * DD;
  const float* brow = tgt + (size_t)(ni * 16 + lm) * DD;

  v8f c = {};
  for (int k = 0; k < DD; k += 4) {
    int kk = k + 2 * half;
    v2f a = *(const v2f*)(arow + kk);
    v2f b = *(const v2f*)(brow + kk);
    c = __builtin_amdgcn_wmma_f32_16x16x4_f32(false, a, false, b,
                                              (short)0, c, false, false);
  }

  // C/D layout: VGPR r -> M = r + 8*(lane/16), N = lane%16
  int gn = ni * 16 + lm;
  float ntg = n2[NN + gn];
#pragma unroll
  for (int r = 0; r < 8; ++r) {
    int gm = mi * 16 + r + 8 * half;
    float d2 = n2[gm] + ntg - 2.0f * c[r];
    d2 = fmaxf(d2, 0.0f);
    sim[gm * NN + gn] = 1.0f / (sqrtf(d2) + 1.0f);
  }
}

// ---------------- Kernel 3: serial selection logic (N=32, trivial work)
__device__ __forceinline__ bool nan_f(float x) { return !(x == x); }

__global__ void msc_finalize(const float* __restrict__ simg,
                             const int* __restrict__ labg,
                             float* __restrict__ out) {
  if (threadIdx.x != 0 || blockIdx.x != 0) return;

  float s[NN][NN];
  int lab[NN];
  for (int i = 0; i < NN; ++i) lab[i] = labg[i];
  for (int i = 0; i < NN; ++i)
    for (int j = 0; j < NN; ++j) s[i][j] = simg[i * NN + j];

  // --- assigned[j]: majority label among top-K rows of column j (stable top-k)
  int assigned[NN];
  for (int j = 0; j < NN; ++j) {
    bool used[NN];
    for (int i = 0; i < NN; ++i) used[i] = false;
    int cnt[UNKC + 1];
    for (int cix = 0; cix <= UNKC; ++cix) cnt[cix] = 0;
    for (int t = 0; t < K_TOP; ++t) {
      int best = -1; float bv = -INFINITY;
      for (int i = 0; i < NN; ++i)
        if (!used[i] && s[i][j] > bv) { bv = s[i][j]; best = i; }
      used[best] = true;
      cnt[lab[best]]++;
    }
    int bi = 0;
    for (int cix = 1; cix <= UNKC; ++cix)
      if (cnt[cix] > cnt[bi]) bi = cix;   // first max wins
    assigned[j] = bi;
  }

  // --- per-column max and sums
  float lmax[NN], colsum[NN];
  for (int j = 0; j < NN; ++j) {
    float m = -INFINITY, cs = 0.f;
    for (int i = 0; i < NN; ++i) { m = fmaxf(m, s[i][j]); cs += s[i][j]; }
    lmax[j] = m; colsum[j] = cs;
  }

  // --- stable ascending argsort of lmax
  int idx_lm[NN];
  {
    bool used[NN];
    for (int j = 0; j < NN; ++j) used[j] = false;
    for (int r = 0; r < NN; ++r) {
      int best = -1;
      for (int j = 0; j < NN; ++j)
        if (!used[j] && (best < 0 || lmax[j] < lmax[best])) best = j;
      used[best] = true; idx_lm[r] = best;
    }
  }

  // --- std_vals over first NU1, stable ascending argsort
  float stdv[NU1];
  for (int t = 0; t < NU1; ++t) stdv[t] = colsum[idx_lm[t]];
  int idx_std[NU1];
  {
    bool used[NU1];
    for (int t = 0; t < NU1; ++t) used[t] = false;
    for (int r = 0; r < NU1; ++r) {
      int best = -1;
      for (int t = 0; t < NU1; ++t)
        if (!used[t] && (best < 0 || stdv[t] < stdv[best])) best = t;
      used[best] = true; idx_std[r] = best;
    }
  }
  int sel[NU2];
  for (int u = 0; u < NU2; ++u) sel[u] = idx_lm[idx_std[u]];
  for (int u = 0; u < NU2; ++u) assigned[sel[u]] = UNKC;

  // --- keep columns not in sel (ascending == stable argsort of 0/1 keys)
  int kidx[NP]; int nk = 0;
  for (int j = 0; j < NN; ++j) {
    bool rem = false;
    for (int u = 0; u < NU2; ++u) if (sel[u] == j) rem = true;
    if (!rem) kidx[nk++] = j;
  }

  // --- per-picked-column score = mean(top3 same) / mean(top3 diff)
  float score[NP];
  for (int t = 0; t < NP; ++t) {
    int col = kidx[t];
    int tl = assigned[col];
    int c_same = 0;
    for (int i = 0; i < NN; ++i) if (lab[i] == tl) c_same++;
    int c_diff = NN - c_same;

    bool used[NN];
    float tops[MR];
    // same-label top-3 (stable)
    for (int i = 0; i < NN; ++i) used[i] = false;
    for (int r = 0; r < MR; ++r) {
      int best = -1; float bv = -INFINITY;
      for (int i = 0; i < NN; ++i)
        if (lab[i] == tl && !used[i] && s[i][col] > bv) { bv = s[i][col]; best = i; }
      if (best >= 0) { used[best] = true; tops[r] = bv; }
      else tops[r] = -INFINITY;
    }
    int take_s = c_same < MR ? c_same : MR;
    float sum_s = 0.f;
    for (int r = 0; r < take_s; ++r) sum_s += tops[r];
    float mean_s = sum_s / (float)take_s;   // 0/0 -> NaN, matches JAX

    // diff-label top-3 (stable)
    for (int i = 0; i < NN; ++i) used[i] = false;
    for (int r = 0; r < MR; ++r) {
      int best = -1; float bv = -INFINITY;
      for (int i = 0; i < NN; ++i)
        if (lab[i] != tl && !used[i] && s[i][col] > bv) { bv = s[i][col]; best = i; }
      if (best >= 0) { used[best] = true; tops[r] = bv; }
      else tops[r] = -INFINITY;
    }
    int take_d = c_diff < MR ? c_diff : MR;
    float sum_d = 0.f;
    for (int r = 0; r < take_d; ++r) sum_d += tops[r];
    float mean_d = sum_d / (float)take_d;

    score[t] = mean_s / mean_d;
  }

  // --- top MU_ scores: stable ascending sort of -score, NaN last
  int tt[MU_];
  {
    bool used[NP];
    for (int t = 0; t < NP; ++t) used[t] = false;
    for (int r = 0; r < MU_; ++r) {
      int best = -1;
      for (int t = 0; t < NP; ++t) {
        if (used[t]) continue;
        if (best < 0) { best = t; continue; }
        float ka = -score[t], kb = -score[best];
        bool less;
        if (nan_f(kb))      less = !nan_f(ka);
        else if (nan_f(ka)) less = false;
        else                less = ka < kb;   // strict -> stable
        if (less) best = t;
      }
      used[best] = true; tt[r] = best;
    }
  }

  // --- final 32x25 contrastive softmax loss
  int flab[FC];
  for (int cix = 0; cix < MU_; ++cix) flab[cix] = assigned[kidx[tt[cix]]];
  for (int cix = 0; cix < NU2; ++cix) flab[MU_ + cix] = UNKC;

  float negsum = 0.f; int vcount = 0;
  for (int i = 0; i < NN; ++i) {
    float fs[FC];
    for (int cix = 0; cix < MU_; ++cix) fs[cix] = s[i][kidx[tt[cix]]];
    for (int cix = 0; cix < NU2; ++cix) fs[MU_ + cix] = s[sel[cix]][i];  // selected_unk

    int msum = 0;
    for (int cix = 0; cix < FC; ++cix) msum += (lab[i] == flab[cix]) ? 1 : 0;
    bool valid = (msum > 0) && (msum < FC);

    float safe[FC]; float mx = -INFINITY;
    for (int cix = 0; cix < FC; ++cix) {
      safe[cix] = valid ? fs[cix] : 0.f;
      mx = fmaxf(mx, safe[cix]);
    }
    float Z = 0.f, cm = 0.f;
    for (int cix = 0; cix < FC; ++cix) {
      float e = expf(safe[cix] - mx);
      Z += e;
      if (lab[i] == flab[cix]) cm += e;
    }
    float contr = cm / Z;
    if (valid) { negsum += -logf(contr); vcount++; }
  }
  float loss = negsum / (float)vcount;   // 0/0 -> NaN like reference

  // --- outputs: loss(1), select_sum_ind(5), assigned(32), sum_line(32)
  out[0] = loss;
  for (int u = 0; u < NU2; ++u) out[1 + u] = (float)sel[u];
  for (int j = 0; j < NN; ++j) out[1 + NU2 + j] = (float)assigned[j];
  for (int j = 0; j < NN; ++j) out[1 + NU2 + NN + j] = colsum[j];
}

extern "C" void kernel_launch(void* const* d_in, const int* in_sizes, int n_in,
                              void* d_out, int out_size, void* d_ws, size_t ws_size,
                              hipStream_t stream) {
  const float* src = (const float*)d_in[0];
  const int*   lab = (const int*)d_in[1];
  const float* tgt = (const float*)d_in[2];
  float* ws  = (float*)d_ws;
  float* n2  = ws;        // 64 floats
  float* sim = ws + 64;   // 32*32 floats

  msc_norms<<<64, 256, 0, stream>>>(src, tgt, n2);
  msc_sim_wmma<<<4, 32, 0, stream>>>(src, tgt, n2, sim);
  msc_finalize<<<1, 1, 0, stream>>>(sim, lab, (float*)d_out);
}